// LinearCRF_87557203296607
// MI455X (gfx1250) — compile-verified
//
#include <hip/hip_runtime.h>
#include <math.h>
#include <stdint.h>

#define DEV __device__ __forceinline__

// Problem constants (from the reference)
constexpr int B = 8192;
constexpr int L = 1024;
constexpr int S = 3;

// Decomposition
constexpr int CHUNKS   = 8;
constexpr int CL       = L / CHUNKS;   // 128 steps per chunk
constexpr int TILE     = 8;            // timesteps staged per LDS tile
constexpr int NT       = CL / TILE;    // 16 tiles per chunk
constexpr int TPB_A    = 128;          // threads per block, chunk kernel
constexpr int TPB_B    = 256;          // threads per block, combine kernel
constexpr int NBLK_B   = B / TPB_B;    // 32 partial sums
constexpr int WS_STRIDE = 12;          // floats per (batch,chunk) record: P[9], gold, pad

// LDS slot per thread per buffer: em 24f | mask 8f | tags 8i | pad 4 = 44 floats (176 B, 16B-aligned)
constexpr int SLOT_F = 44;
constexpr float NEG = -1.0e30f;

// ---------------- CDNA5 async global->LDS path ----------------
#if defined(__HIP_DEVICE_COMPILE__) && __has_builtin(__builtin_amdgcn_global_load_async_to_lds_b128)
#define HAS_ASYNC 1
#else
#define HAS_ASYNC 0
#endif

typedef int v4i __attribute__((vector_size(16)));
#if HAS_ASYNC
typedef __attribute__((address_space(1))) v4i gv4i_t;
typedef __attribute__((address_space(3))) v4i lv4i_t;
#endif

DEV void copy16(const void* g, void* l) {
#if HAS_ASYNC
  __builtin_amdgcn_global_load_async_to_lds_b128((gv4i_t*)g, (lv4i_t*)l, 0, 0);
#else
  *(float4*)l = *(const float4*)g;   // synchronous fallback
#endif
}

template <int N>
DEV void wait_async_le() {
#if HAS_ASYNC
#if __has_builtin(__builtin_amdgcn_s_wait_asynccnt)
  __builtin_amdgcn_s_wait_asynccnt(N);
#else
  asm volatile("s_wait_asynccnt %0" ::"n"(N) : "memory");
#endif
#endif
}

// Stage one 8-step tile (96B em, 32B mask, 32B tags) into this thread's LDS slot.
DEV void prefetch_tile(const float* em_t, const float* mask_t, const int* tags_t, float* slot) {
  const char* ge = (const char*)em_t;
  char*       ls = (char*)slot;
  copy16(ge + 0,  ls + 0);
  copy16(ge + 16, ls + 16);
  copy16(ge + 32, ls + 32);
  copy16(ge + 48, ls + 48);
  copy16(ge + 64, ls + 64);
  copy16(ge + 80, ls + 80);
  const char* gm = (const char*)mask_t;
  copy16(gm + 0,  ls + 96);
  copy16(gm + 16, ls + 112);
  const char* gt = (const char*)tags_t;
  copy16(gt + 0,  ls + 128);
  copy16(gt + 16, ls + 144);
}
constexpr int ASYNC_PER_TILE = 10;

DEV float lse3(float x, float y, float z) {
  float m = fmaxf(x, fmaxf(y, z));
  return m + __logf(__expf(x - m) + __expf(y - m) + __expf(z - m));
}

// ---------------- Kernel A: per-(batch,chunk) 3x3 log-semiring product + gold partial ----------
__global__ __launch_bounds__(TPB_A) void crf_chunk_kernel(
    const float* __restrict__ em, const float* __restrict__ mask,
    const float* __restrict__ trans, const int* __restrict__ tags,
    float* __restrict__ ws)
{
  __shared__ float lds[2][TPB_A][SLOT_F];
  __shared__ float sT[12];   // transitions for dynamic (prev,cur) lookup (avoid scratch)

  const int tid   = threadIdx.x;
  const int b     = blockIdx.x * TPB_A + tid;
  const int chunk = blockIdx.y;
  const int t0c   = chunk * CL;

  if (tid < 9) sT[tid] = trans[tid];
  __syncthreads();

  const float* em_b   = em   + (size_t)b * (L * S);
  const float* mask_b = mask + (size_t)b * L;
  const int*   tags_b = tags + (size_t)b * L;

  // exp(transitions) for linear-domain matvec (entries are small; no overflow)
  float ET[9];
#pragma unroll
  for (int i = 0; i < 9; ++i) ET[i] = __expf(trans[i]);

  // R = identity in the log semiring
  float R[9];
#pragma unroll
  for (int i = 0; i < 9; ++i) R[i] = ((i % 4) == 0) ? 0.0f : NEG;

  float gold = 0.0f;
  int prev = (chunk > 0) ? tags_b[t0c - 1] : 0;

  prefetch_tile(em_b + (size_t)t0c * S, mask_b + t0c, tags_b + t0c, &lds[0][tid][0]);

  for (int tt = 0; tt < NT; ++tt) {
    if (tt + 1 < NT) {
      const int tn = t0c + (tt + 1) * TILE;
      prefetch_tile(em_b + (size_t)tn * S, mask_b + tn, tags_b + tn, &lds[(tt + 1) & 1][tid][0]);
      wait_async_le<ASYNC_PER_TILE>();   // current tile's 10 loads complete (in-order)
    } else {
      wait_async_le<0>();
    }
    const float* slot   = &lds[tt & 1][tid][0];
    const int*   slot_i = (const int*)slot;

#pragma unroll
    for (int k = 0; k < TILE; ++k) {
      const int   t  = t0c + tt * TILE + k;
      const float e0 = slot[3 * k + 0];
      const float e1 = slot[3 * k + 1];
      const float e2 = slot[3 * k + 2];
      const float mk = slot[24 + k];
      const int   tg = slot_i[32 + k];
      const float eg = (tg == 0) ? e0 : ((tg == 1) ? e1 : e2);

      if (t == 0) {                       // only chunk 0, very first step
        gold = eg * mk;                   // em_score[:,0] * mask[:,0]
      } else {
        gold += (sT[prev * 3 + tg] + eg) * mk;
        if (mk > 0.0f) {
          // R <- R (x) M_t,  M_t[i][j] = trans[i][j] + em_t[j]
#pragma unroll
          for (int i = 0; i < 3; ++i) {
            const float a0 = R[i * 3 + 0], a1 = R[i * 3 + 1], a2 = R[i * 3 + 2];
            const float m  = fmaxf(a0, fmaxf(a1, a2));
            const float x0 = __expf(a0 - m), x1 = __expf(a1 - m), x2 = __expf(a2 - m);
            const float s0 = x0 * ET[0] + x1 * ET[3] + x2 * ET[6];
            const float s1 = x0 * ET[1] + x1 * ET[4] + x2 * ET[7];
            const float s2 = x0 * ET[2] + x1 * ET[5] + x2 * ET[8];
            R[i * 3 + 0] = m + __logf(s0) + e0;
            R[i * 3 + 1] = m + __logf(s1) + e1;
            R[i * 3 + 2] = m + __logf(s2) + e2;
          }
        }
      }
      prev = tg;
    }
  }

  float* rec = ws + ((size_t)b * CHUNKS + chunk) * WS_STRIDE;
#pragma unroll
  for (int i = 0; i < 9; ++i) rec[i] = R[i];
  rec[9] = gold;
}

// ---------------- Kernel B: combine chunks per batch, block-reduce (logZ - gold) --------------
__global__ __launch_bounds__(TPB_B) void crf_combine_kernel(
    const float* __restrict__ em, const float* __restrict__ ws_in,
    float* __restrict__ part)
{
  const int tid = threadIdx.x;
  const int b   = blockIdx.x * TPB_B + tid;

  const float* em0 = em + (size_t)b * (L * S);
  float a0 = em0[0], a1 = em0[1], a2 = em0[2];   // alpha0 = emissions[:,0,:]
  float gold = 0.0f;

  for (int c = 0; c < CHUNKS; ++c) {
    const float* rec = ws_in + ((size_t)b * CHUNKS + c) * WS_STRIDE;
    float P[9];
#pragma unroll
    for (int i = 0; i < 9; ++i) P[i] = rec[i];
    gold += rec[9];
    const float n0 = lse3(a0 + P[0], a1 + P[3], a2 + P[6]);
    const float n1 = lse3(a0 + P[1], a1 + P[4], a2 + P[7]);
    const float n2 = lse3(a0 + P[2], a1 + P[5], a2 + P[8]);
    a0 = n0; a1 = n1; a2 = n2;
  }

  const float val = lse3(a0, a1, a2) - gold;     // logZ - gold

  __shared__ float red[TPB_B];
  red[tid] = val;
  __syncthreads();
  for (int s = TPB_B / 2; s > 0; s >>= 1) {
    if (tid < s) red[tid] += red[tid + s];
    __syncthreads();
  }
  if (tid == 0) part[blockIdx.x] = red[0];
}

// ---------------- Kernel C: deterministic final mean ----------------
__global__ void crf_finalize_kernel(const float* __restrict__ part, float* __restrict__ out) {
  if (threadIdx.x == 0 && blockIdx.x == 0) {
    float s = 0.0f;
    for (int i = 0; i < NBLK_B; ++i) s += part[i];
    out[0] = s / (float)B;
  }
}

// ---------------- Launch ----------------
extern "C" void kernel_launch(void* const* d_in, const int* in_sizes, int n_in,
                              void* d_out, int out_size, void* d_ws, size_t ws_size,
                              hipStream_t stream) {
  (void)in_sizes; (void)n_in; (void)out_size; (void)ws_size;
  const float* em    = (const float*)d_in[0];
  const float* mask  = (const float*)d_in[1];
  const float* trans = (const float*)d_in[2];
  const int*   tags  = (const int*)d_in[3];
  float* ws   = (float*)d_ws;
  float* out  = (float*)d_out;
  float* part = ws + (size_t)B * CHUNKS * WS_STRIDE;   // 32 block partials after records (~3 MB)

  crf_chunk_kernel<<<dim3(B / TPB_A, CHUNKS), dim3(TPB_A), 0, stream>>>(em, mask, trans, tags, ws);
  crf_combine_kernel<<<dim3(B / TPB_B), dim3(TPB_B), 0, stream>>>(em, ws, part);
  crf_finalize_kernel<<<dim3(1), dim3(64), 0, stream>>>(part, out);
}